// QAEEncoder_47777216201062
// MI455X (gfx1250) — compile-verified
//
#include <hip/hip_runtime.h>
#include <math.h>

typedef float v2f __attribute__((ext_vector_type(2)));
typedef float v8f __attribute__((ext_vector_type(8)));

// ---------------------------------------------------------------------------
// Prep kernel: build 12 fused gate matrices (3 stages x 4 qubit-groups),
// each an 8x8 complex unitary stored as a 16x16 real matrix (row-major)
// in d_ws.  Stage gates (all single-qubit gates between CNOT rings commute):
//   stage0: U = RZ(w[10+q]) * RY(w[q])
//   stage1: U = RZ(w[40+q]) * RY(w[20+q]+w[30+q])
//   stage2: U = RY(w[50+q])
// Groups are triples of FLAT bits (JAX reshape: qubit q <-> flat bit 9-q):
//   g0: flat 0-2 (qubits 9,8,7)   g1: flat 3-5 (qubits 6,5,4)
//   g2: flat 6-8 (qubits 3,2,1)   g3: flat 7-9, identity on 7,8, gate on 9 (qubit 0)
// ---------------------------------------------------------------------------
__device__ __forceinline__ void stage_gate(const float* w, int s, int q,
                                           float ur[2][2], float ui[2][2]) {
  float ry, rz;
  if (s == 0)      { ry = w[q];               rz = w[10 + q]; }
  else if (s == 1) { ry = w[20 + q] + w[30 + q]; rz = w[40 + q]; }
  else             { ry = w[50 + q];          rz = 0.0f; }
  float c  = cosf(ry * 0.5f), sn = sinf(ry * 0.5f);
  float er = cosf(rz * 0.5f), ei = sinf(rz * 0.5f);
  // U = RZ * RY ; RZ = diag(conj(e), e), e = er + i*ei
  ur[0][0] =  c * er;  ui[0][0] = -c * ei;
  ur[0][1] = -sn * er; ui[0][1] =  sn * ei;
  ur[1][0] =  sn * er; ui[1][0] =  sn * ei;
  ur[1][1] =  c * er;  ui[1][1] =  c * ei;
}

__global__ void qae_prep(const float* __restrict__ w, float* __restrict__ mats) {
  int m = threadIdx.x;
  if (m >= 12) return;
  int s = m >> 2, g = m & 3;
  int a = (g < 3) ? 3 * g : 7;   // flat-bit base of this group
  float ur[3][2][2], ui[3][2][2];
  for (int l = 0; l < 3; ++l) {
    if (g == 3 && l < 2) {       // identity on flat bits 7,8 (handled by g2)
      ur[l][0][0] = 1.f; ur[l][0][1] = 0.f; ur[l][1][0] = 0.f; ur[l][1][1] = 1.f;
      ui[l][0][0] = 0.f; ui[l][0][1] = 0.f; ui[l][1][0] = 0.f; ui[l][1][1] = 0.f;
    } else {
      int q = 9 - (a + l);       // flat bit -> qubit
      stage_gate(w, s, q, ur[l], ui[l]);
    }
  }
  float* Mp = mats + m * 256;
  for (int r = 0; r < 8; ++r) {
    for (int c = 0; c < 8; ++c) {
      float g0r = ur[0][r & 1][c & 1],             g0i = ui[0][r & 1][c & 1];
      float g1r = ur[1][(r >> 1) & 1][(c >> 1) & 1], g1i = ui[1][(r >> 1) & 1][(c >> 1) & 1];
      float g2r = ur[2][(r >> 2) & 1][(c >> 2) & 1], g2i = ui[2][(r >> 2) & 1][(c >> 2) & 1];
      float pr = g0r * g1r - g0i * g1i, pi = g0r * g1i + g0i * g1r;
      float qr = pr * g2r - pi * g2i,   qi = pr * g2i + pi * g2r;
      // real 2x2-block encoding of the complex entry
      Mp[(2 * r) * 16 + 2 * c]         =  qr;
      Mp[(2 * r) * 16 + 2 * c + 1]     = -qi;
      Mp[(2 * r + 1) * 16 + 2 * c]     =  qi;
      Mp[(2 * r + 1) * 16 + 2 * c + 1] =  qr;
    }
  }
}

// ---------------------------------------------------------------------------
// Main kernel: one wave per sample, state in LDS (float2 = complex amp).
// Each gate-group application = 16x16 real matmul over 128 state columns,
// done as 8 tiles of 4 chained v_wmma_f32_16x16x4_f32.
// ---------------------------------------------------------------------------
__device__ __forceinline__ v8f wmma_step(float2 A, v2f b, v8f c) {
  v2f a; a.x = A.x; a.y = A.y;
  return __builtin_amdgcn_wmma_f32_16x16x4_f32(false, a, false, b, (short)0, c,
                                               false, false);
}

__device__ __forceinline__ v2f ld_amp(const float2* src, const unsigned short* perm,
                                      bool use_perm, int idx) {
  int ph = use_perm ? (int)perm[idx] : idx;
  float2 v = src[ph];
  v2f r; r.x = v.x; r.y = v.y; return r;
}

__device__ __forceinline__ void apply_stage(const float2* src, float2* dst,
                                            const float* Ms,
                                            const unsigned short* perm,
                                            bool use_perm, int lane) {
  const int row = lane & 15;   // A row (=output real index) / B,D column
  const int hi  = lane >> 4;   // lane half selects K/M sub-block
  const int bases[4] = {0, 3, 6, 7};
#pragma unroll
  for (int g = 0; g < 4; ++g) {
    const float* M = Ms + g * 256;
    // A operand, 16x16 in 4 K-chunks of 16x4 (layout per ISA 7.12.2)
    float2 A0 = *(const float2*)(M + row * 16 + 0  + 2 * hi);
    float2 A1 = *(const float2*)(M + row * 16 + 4  + 2 * hi);
    float2 A2 = *(const float2*)(M + row * 16 + 8  + 2 * hi);
    float2 A3 = *(const float2*)(M + row * 16 + 12 + 2 * hi);
    const int a  = bases[g];
    const int lm = (1 << a) - 1;
    for (int t = 0; t < 8; ++t) {
      int c    = t * 16 + row;                       // column = non-group bits
      int bidx = (c & lm) | ((c >> a) << (a + 3));   // insert 3-bit subspace slot
      // B operand: lane holds (re,im) of amps j = 2k'+hi of its column
      v2f b0 = ld_amp(src, perm, use_perm, bidx | ((hi + 0) << a));
      v2f b1 = ld_amp(src, perm, use_perm, bidx | ((hi + 2) << a));
      v2f b2 = ld_amp(src, perm, use_perm, bidx | ((hi + 4) << a));
      v2f b3 = ld_amp(src, perm, use_perm, bidx | ((hi + 6) << a));
      v8f d = {0.f, 0.f, 0.f, 0.f, 0.f, 0.f, 0.f, 0.f};
      d = wmma_step(A0, b0, d);
      d = wmma_step(A1, b1, d);
      d = wmma_step(A2, b2, d);
      d = wmma_step(A3, b3, d);
      // D: lane (lo) holds amps 0..3 of its column, lane (hi) amps 4..7
      int js = hi * 4;
      dst[bidx | ((js + 0) << a)] = make_float2(d[0], d[1]);
      dst[bidx | ((js + 1) << a)] = make_float2(d[2], d[3]);
      dst[bidx | ((js + 2) << a)] = make_float2(d[4], d[5]);
      dst[bidx | ((js + 3) << a)] = make_float2(d[6], d[7]);
    }
  }
}

__global__ __launch_bounds__(64) void qae_main(const float* __restrict__ x,
                                               const float* __restrict__ mats,
                                               float* __restrict__ out, int B) {
  __shared__ float2 sbuf[2][2][1024];      // 2 waves x double buffer x 1024 amps
  __shared__ unsigned short perm[1024];    // CNOT-ring gather permutation
  const int tid = threadIdx.x;

  // warm L2 with this block's input rows while building the perm table
  {
    int bp = blockIdx.x * 2 + (tid >> 5);
    if (bp < B) __builtin_prefetch(x + (size_t)bp * 768 + (tid & 31) * 16, 0, 0);
  }
  // sigma(i): source index (pre-ring) feeding post-ring index i.
  // CNOT(q -> q+1 mod 10) applied q=0..9; flat bits: cb=9-q, tb=9-((q+1)%10).
  for (int i = tid; i < 1024; i += 64) {
    int v = i;
    for (int q = 9; q >= 0; --q) {
      int cb = 9 - q;
      int tb = 9 - ((q + 1) % 10);
      v ^= ((v >> cb) & 1) << tb;
    }
    perm[i] = (unsigned short)v;
  }
  __syncthreads();

  const int w = tid >> 5, lane = tid & 31;
  const int b = blockIdx.x * 2 + w;
  if (b >= B) return;
  const float* xb = x + (size_t)b * 768;
  float2* buf0 = sbuf[w][0];
  float2* buf1 = sbuf[w][1];

  // ---- load row, L2-normalize, write complex state (im = 0), pad to 1024
  float4 vv[6];
  float ss = 0.f;
#pragma unroll
  for (int u = 0; u < 6; ++u) {
    vv[u] = ((const float4*)xb)[lane + 32 * u];
    ss += vv[u].x * vv[u].x + vv[u].y * vv[u].y + vv[u].z * vv[u].z + vv[u].w * vv[u].w;
  }
#pragma unroll
  for (int m = 16; m >= 1; m >>= 1) ss += __shfl_xor(ss, m, 32);
  float inv = 1.0f / fmaxf(sqrtf(ss), 1e-8f);
#pragma unroll
  for (int u = 0; u < 6; ++u) {
    int base = (lane + 32 * u) * 4;
    buf0[base + 0] = make_float2(vv[u].x * inv, 0.f);
    buf0[base + 1] = make_float2(vv[u].y * inv, 0.f);
    buf0[base + 2] = make_float2(vv[u].z * inv, 0.f);
    buf0[base + 3] = make_float2(vv[u].w * inv, 0.f);
  }
  for (int i = 768 + lane; i < 1024; i += 32) buf0[i] = make_float2(0.f, 0.f);

  // ---- 3 fused single-qubit stages; CNOT rings folded into gathers
  apply_stage(buf0, buf1, mats + 0,    perm, false, lane);
  apply_stage(buf1, buf0, mats + 1024, perm, true,  lane);
  apply_stage(buf0, buf1, mats + 2048, perm, true,  lane);

  // ---- expectations <X>,<Y>,<Z> on qubits 0..5 (flat bit p = 9-q)
#pragma unroll
  for (int q = 0; q < 6; ++q) {
    const int p  = 9 - q;
    const int pm = (1 << p) - 1;
    float xa = 0.f, ya = 0.f, za = 0.f;
    for (int u = 0; u < 16; ++u) {
      int pp = lane + 32 * u;                        // 512 pairs
      int i0 = ((pp >> p) << (p + 1)) | (pp & pm);   // bit p = 0
      int i1 = i0 | (1 << p);                        // bit p = 1
      float2 s0 = buf1[i0], s1 = buf1[i1];
      xa += s0.x * s1.x + s0.y * s1.y;               // Re(conj(s0)*s1)
      ya += s0.x * s1.y - s0.y * s1.x;               // Im(conj(s0)*s1)
      za += (s0.x * s0.x + s0.y * s0.y) - (s1.x * s1.x + s1.y * s1.y);
    }
#pragma unroll
    for (int m = 16; m >= 1; m >>= 1) {
      xa += __shfl_xor(xa, m, 32);
      ya += __shfl_xor(ya, m, 32);
      za += __shfl_xor(za, m, 32);
    }
    if (lane == 0) {
      out[(size_t)b * 18 + q]      = 2.f * xa;
      out[(size_t)b * 18 + 6 + q]  = 2.f * ya;
      out[(size_t)b * 18 + 12 + q] = za;
    }
  }
}

extern "C" void kernel_launch(void* const* d_in, const int* in_sizes, int n_in,
                              void* d_out, int out_size, void* d_ws, size_t ws_size,
                              hipStream_t stream) {
  const float* x   = (const float*)d_in[0];
  const float* wts = (const float*)d_in[1];
  float* out  = (float*)d_out;
  float* mats = (float*)d_ws;              // 12 * 256 floats = 12 KB
  int B = in_sizes[0] / 768;
  qae_prep<<<1, 32, 0, stream>>>(wts, mats);
  int nblk = (B + 1) / 2;                  // 2 samples (waves) per block
  qae_main<<<nblk, 64, 0, stream>>>(x, mats, out, B);
}